// GAT_23613730193923
// MI455X (gfx1250) — compile-verified
//
#include <hip/hip_runtime.h>
#include <math.h>
#include <stdint.h>

// ---------------------------------------------------------------- constants
#define BATCH 1024
#define NPTS  80          // PAST + FUT
#define PAST  64
#define FUT   16
#define DM    26
#define OUTP  51
#define OPAD  52          // 51 padded to multiple of 4
#define OFEAT 50          // O
#define HID1  64
#define HID2  64
#define HIDK  64          // Wk inner dim
#define NEGV  (-1e30f)

typedef float v2f __attribute__((ext_vector_type(2)));
typedef float v8f __attribute__((ext_vector_type(8)));

// ---------------------------------------------------------------- wmma helpers
__device__ __forceinline__ v8f wmma4(v2f a, v2f b, v8f c) {
  // D = A(16x4,f32) * B(4x16,f32) + C(16x16,f32)
  return __builtin_amdgcn_wmma_f32_16x16x4_f32(false, a, false, b, (short)0, c,
                                               false, false);
}

__device__ __forceinline__ v8f splat8(float v) {
  v8f r;
#pragma unroll
  for (int i = 0; i < 8; ++i) r[i] = v;
  return r;
}

// A fragment: 16x4 tile at (m0,k0) of row-major matrix with leading dim ld.
// lanes 0-15: K = k0,k0+1 ; lanes 16-31: K = k0+2,k0+3 (ISA 7.12.2 layout)
__device__ __forceinline__ v2f afrag(const float* base, int ld, int m0, int k0) {
  int lane = threadIdx.x & 31;
  int row  = m0 + (lane & 15);
  int k    = k0 + ((lane >> 4) << 1);
  v2f a;
  a[0] = base[row * ld + k];
  a[1] = base[row * ld + k + 1];
  return a;
}

// B fragment from row-major B[k][n] with leading dim ld at (k0,n0).
__device__ __forceinline__ v2f bfrag(const float* base, int ld, int k0, int n0) {
  int lane = threadIdx.x & 31;
  int col  = n0 + (lane & 15);
  int k    = k0 + ((lane >> 4) << 1);
  v2f b;
  b[0] = base[(k + 0) * ld + col];
  b[1] = base[(k + 1) * ld + col];
  return b;
}

// ---------------------------------------------------------------- async global->LDS
// 16B-per-lane async copy (GLOBAL_LOAD_ASYNC_TO_LDS_B128), tracked by ASYNCcnt.
typedef int v4i_vs __attribute__((vector_size(16)));
typedef __attribute__((address_space(1))) v4i_vs* gbl_v4i_p;   // global (AS1)
typedef __attribute__((address_space(3))) v4i_vs* lds_v4i_p;   // LDS   (AS3)

__device__ __forceinline__ void async_g2l_b128(const float* g, float* l) {
#if __has_builtin(__builtin_amdgcn_global_load_async_to_lds_b128)
  __builtin_amdgcn_global_load_async_to_lds_b128(
      (gbl_v4i_p)(uintptr_t)g, (lds_v4i_p)(uintptr_t)(unsigned)(uintptr_t)l,
      0, 0);
#else
  unsigned lds = (unsigned)(uintptr_t)l;   // low 32 bits = LDS offset
  asm volatile("global_load_async_to_lds_b128 %0, %1, off"
               :: "v"(lds), "v"(g) : "memory");
#endif
}

__device__ __forceinline__ void wait_async() {
#if __has_builtin(__builtin_amdgcn_s_wait_asynccnt)
  __builtin_amdgcn_s_wait_asynccnt(0);
#else
  asm volatile("s_wait_asynccnt 0" ::: "memory");
#endif
}

// ---------------------------------------------------------------- kernel 1: precompute
__global__ void k_precompute(const float* __restrict__ Mg,
                             const float* __restrict__ Wk,
                             const float* __restrict__ smoothing,
                             float* __restrict__ Aout,
                             float* __restrict__ An11,
                             float* __restrict__ An22,
                             float* __restrict__ Atm,
                             float* __restrict__ theta,
                             float* __restrict__ scale) {
  __shared__ float As[NPTS * NPTS];
  __shared__ float D1[PAST];
  __shared__ float D2[FUT];
  const int tid = threadIdx.x;

  // rows of A = masked softmax(relu(M M^T)) over q<p, zero elsewhere
  for (int p = tid; p < NPTS; p += blockDim.x) {
    float row[NPTS];
    float mx = NEGV;
    for (int q = 0; q < NPTS; ++q) {
      if (q < p) {
        float d = 0.f;
        for (int k = 0; k < DM; ++k) d += Mg[p * DM + k] * Mg[q * DM + k];
        row[q] = fmaxf(d, 0.f);
        mx = fmaxf(mx, row[q]);
      } else {
        row[q] = NEGV;
      }
    }
    float sum = 0.f;
    for (int q = 0; q < p; ++q) { row[q] = expf(row[q] - mx); sum += row[q]; }
    float inv = (p > 0) ? (1.f / sum) : 0.f;
    for (int q = 0; q < NPTS; ++q) {
      float v = (q < p) ? row[q] * inv : 0.f;
      As[p * NPTS + q]   = v;
      Aout[p * NPTS + q] = v;
    }
  }
  __syncthreads();

  if (tid < PAST) {
    int cnt = 0;
    for (int q = 0; q < PAST; ++q) cnt += (As[tid * NPTS + q] > 1e-15f) ? 1 : 0;
    D1[tid] = rsqrtf(1.f + (float)cnt);
  } else if (tid < PAST + FUT) {
    int i = tid - PAST, cnt = 0;
    for (int q = 0; q < FUT; ++q)
      cnt += (As[(PAST + i) * NPTS + PAST + q] > 1e-15f) ? 1 : 0;
    D2[i] = rsqrtf(1.f + (float)cnt);
  }
  __syncthreads();

  for (int e = tid; e < PAST * PAST; e += blockDim.x) {
    int i = e >> 6, q = e & 63;
    An11[e] = D1[i] * As[i * NPTS + q] * D1[q];
  }
  for (int e = tid; e < FUT * FUT; e += blockDim.x) {
    int i = e >> 4, q = e & 15;
    An22[e] = D2[i] * As[(PAST + i) * NPTS + PAST + q] * D2[q];
  }

  // At = softmax_i( where(A[64+j][i]!=0, ., NEG) ); mask = At >= 0.004
  if (tid < FUT) {
    int j = tid;
    float v[PAST], mx = NEGV;
    for (int i = 0; i < PAST; ++i) {
      float a = As[(PAST + j) * NPTS + i];
      v[i] = (a != 0.f) ? a : NEGV;
      mx = fmaxf(mx, v[i]);
    }
    float sum = 0.f;
    for (int i = 0; i < PAST; ++i) { v[i] = expf(v[i] - mx); sum += v[i]; }
    float inv = 1.f / sum;
    for (int i = 0; i < PAST; ++i)
      Atm[j * PAST + i] = (v[i] * inv >= 0.004f) ? 1.f : 0.f;
  }

  // theta = Wk Wk^T, zero-padded to 52x52
  for (int e = tid; e < OPAD * OPAD; e += blockDim.x) {
    int p = e / OPAD, q = e % OPAD;
    float v = 0.f;
    if (p < OFEAT && q < OFEAT)
      for (int k = 0; k < HIDK; ++k) v += Wk[p * HIDK + k] * Wk[q * HIDK + k];
    theta[e] = v;
  }

  if (tid == 0) {
    float s   = smoothing[0];
    float sig = 1.f / (1.f + expf(-s));
    scale[0]  = -0.5f / (sig * 0.01f);
  }
}

// ---------------------------------------------------------------- kernel 2: embed + MLP
__global__ void k_mlp(const float* __restrict__ data,
                      const float* __restrict__ emb0,
                      const float* __restrict__ emb1,
                      const float* __restrict__ emb2,
                      const float* __restrict__ w1, const float* __restrict__ b1,
                      const float* __restrict__ w2, const float* __restrict__ b2,
                      const float* __restrict__ w3, const float* __restrict__ b3,
                      float* __restrict__ xout) {
  __shared__ float bufA[2][16 * 128];
  __shared__ float bufB[2][16 * 128];
  const int wave = threadIdx.x >> 5;
  const int lane = threadIdx.x & 31;
  const int r0   = (blockIdx.x * 2 + wave) * 16;   // first flat row of this tile
  float* Abuf = bufA[wave];
  float* Bbuf = bufB[wave];

  __builtin_prefetch(w1, 0, 3);
  __builtin_prefetch(w2, 0, 3);
  __builtin_prefetch(w3, 0, 3);

  // gather embeddings + continuous features into Bbuf[16][0..51] (stride 128)
  for (int e = lane; e < 16 * OPAD; e += 32) {
    int r = e / OPAD, c = e % OPAD;
    const float* drow = data + (size_t)(r0 + r) * 6;
    float v;
    if (c < 32)       v = emb0[(int)drow[0] * 32 + c];
    else if (c < 40)  v = emb1[(int)drow[1] * 8 + (c - 32)];
    else if (c < 48)  v = emb2[(int)drow[2] * 8 + (c - 40)];
    else if (c < 51)  v = drow[3 + (c - 48)];
    else              v = 0.f;
    Bbuf[r * 128 + c] = v;
  }
  __syncthreads();

  // layer 1: (16x52) x (52x128), bias+relu -> Abuf
  for (int nt = 0; nt < 8; ++nt) {
    int col = nt * 16 + (lane & 15);
    v8f c = splat8(b1[col]);
    for (int ks = 0; ks < 13; ++ks) {
      v2f a = afrag(Bbuf, 128, 0, ks * 4);
      int k = ks * 4 + ((lane >> 4) << 1);
      v2f b;
      b[0] = (k + 0 < OUTP) ? w1[col * OUTP + k + 0] : 0.f;
      b[1] = (k + 1 < OUTP) ? w1[col * OUTP + k + 1] : 0.f;
      c = wmma4(a, b, c);
    }
#pragma unroll
    for (int r = 0; r < 8; ++r) {
      int m = r + ((lane >> 4) << 3);
      Abuf[m * 128 + col] = fmaxf(c[r], 0.f);
    }
  }
  __syncthreads();

  // layer 2: (16x128) x (128x128), bias+relu -> Bbuf
  for (int nt = 0; nt < 8; ++nt) {
    int col = nt * 16 + (lane & 15);
    v8f c = splat8(b2[col]);
    for (int ks = 0; ks < 32; ++ks) {
      v2f a = afrag(Abuf, 128, 0, ks * 4);
      int k = ks * 4 + ((lane >> 4) << 1);
      v2f b;
      b[0] = w2[col * 128 + k + 0];
      b[1] = w2[col * 128 + k + 1];
      c = wmma4(a, b, c);
    }
#pragma unroll
    for (int r = 0; r < 8; ++r) {
      int m = r + ((lane >> 4) << 3);
      Bbuf[m * 128 + col] = fmaxf(c[r], 0.f);
    }
  }
  __syncthreads();

  // layer 3: (16x128) x (128x50), bias -> x[:,0..49]; col50 = data[...,5]; col51 = 0
  for (int nt = 0; nt < 4; ++nt) {
    int col = nt * 16 + (lane & 15);
    v8f c = splat8((col < OFEAT) ? b3[col] : 0.f);
    for (int ks = 0; ks < 32; ++ks) {
      v2f a = afrag(Bbuf, 128, 0, ks * 4);
      int k = ks * 4 + ((lane >> 4) << 1);
      v2f b;
      b[0] = (col < OFEAT) ? w3[col * 128 + k + 0] : 0.f;
      b[1] = (col < OFEAT) ? w3[col * 128 + k + 1] : 0.f;
      c = wmma4(a, b, c);
    }
    if (col < OFEAT) {
#pragma unroll
      for (int r = 0; r < 8; ++r) {
        int m = r + ((lane >> 4) << 3);
        xout[(size_t)(r0 + m) * OPAD + col] = c[r];
      }
    }
  }
  if (lane < 16) {
    int row = r0 + lane;
    xout[(size_t)row * OPAD + 50] = data[(size_t)row * 6 + 5];
  } else {
    int row = r0 + (lane - 16);
    xout[(size_t)row * OPAD + 51] = 0.f;   // keep padded rows raw-copyable
  }
}

// ---------------------------------------------------------------- kernel 3: GCNN #1
__global__ void k_gcnn1(const float* __restrict__ xin,
                        const float* __restrict__ An11,
                        const float* __restrict__ Wg1a,
                        const float* __restrict__ Wg1b,
                        float* __restrict__ xpout) {
  __shared__ __align__(16) float Xs[PAST * OPAD];
  __shared__ float Ts[PAST * HID1];
  __shared__ float Us[PAST * HID1];
  const int b    = blockIdx.x;
  const int tid  = threadIdx.x;
  const int wave = tid >> 5;
  const int lane = tid & 31;
  const float* xb = xin + (size_t)b * NPTS * OPAD;

  // async stage the whole 64x52 tile (col 51 is zero in global x)
  for (int e = tid * 4; e < PAST * OPAD; e += 64 * 4)
    async_g2l_b128(xb + e, Xs + e);
  wait_async();
  __syncthreads();

  // stage 1: Ts = X * Wg1a^T   (64x52)x(52x64)
  for (int t = wave; t < 16; t += 2) {
    int mi = t >> 2, ni = t & 3;
    int col = ni * 16 + (lane & 15);
    v8f c = splat8(0.f);
    for (int ks = 0; ks < 13; ++ks) {
      v2f a = afrag(Xs, OPAD, mi * 16, ks * 4);
      int k = ks * 4 + ((lane >> 4) << 1);
      v2f bb;
      bb[0] = (k + 0 < OUTP) ? Wg1a[col * OUTP + k + 0] : 0.f;
      bb[1] = (k + 1 < OUTP) ? Wg1a[col * OUTP + k + 1] : 0.f;
      c = wmma4(a, bb, c);
    }
#pragma unroll
    for (int r = 0; r < 8; ++r)
      Ts[(mi * 16 + r + ((lane >> 4) << 3)) * HID1 + col] = c[r];
  }
  __syncthreads();

  // stage 2: Us = relu(An11 * Ts)
  for (int t = wave; t < 16; t += 2) {
    int mi = t >> 2, ni = t & 3;
    int col = ni * 16 + (lane & 15);
    v8f c = splat8(0.f);
    for (int ks = 0; ks < 16; ++ks) {
      v2f a  = afrag(An11, PAST, mi * 16, ks * 4);
      v2f bb = bfrag(Ts, HID1, ks * 4, ni * 16);
      c = wmma4(a, bb, c);
    }
#pragma unroll
    for (int r = 0; r < 8; ++r)
      Us[(mi * 16 + r + ((lane >> 4) << 3)) * HID1 + col] = fmaxf(c[r], 0.f);
  }
  __syncthreads();

  // stage 3: Ts = Us * Wg1b^T  (64x64)x(64x50 pad 64)
  for (int t = wave; t < 16; t += 2) {
    int mi = t >> 2, ni = t & 3;
    int col = ni * 16 + (lane & 15);
    v8f c = splat8(0.f);
    for (int ks = 0; ks < 16; ++ks) {
      v2f a = afrag(Us, HID1, mi * 16, ks * 4);
      int k = ks * 4 + ((lane >> 4) << 1);
      v2f bb;
      bb[0] = (col < OFEAT) ? Wg1b[col * HID1 + k + 0] : 0.f;
      bb[1] = (col < OFEAT) ? Wg1b[col * HID1 + k + 1] : 0.f;
      c = wmma4(a, bb, c);
    }
#pragma unroll
    for (int r = 0; r < 8; ++r)
      Ts[(mi * 16 + r + ((lane >> 4) << 3)) * HID1 + col] = c[r];
  }
  __syncthreads();

  // stage 4: xp = An11 * Ts -> global (stride 52, cols < 50)
  for (int t = wave; t < 16; t += 2) {
    int mi = t >> 2, ni = t & 3;
    int col = ni * 16 + (lane & 15);
    v8f c = splat8(0.f);
    for (int ks = 0; ks < 16; ++ks) {
      v2f a  = afrag(An11, PAST, mi * 16, ks * 4);
      v2f bb = bfrag(Ts, HID1, ks * 4, ni * 16);
      c = wmma4(a, bb, c);
    }
    if (col < OFEAT) {
#pragma unroll
      for (int r = 0; r < 8; ++r) {
        int m = mi * 16 + r + ((lane >> 4) << 3);
        xpout[((size_t)b * PAST + m) * OPAD + col] = c[r];
      }
    }
  }
}

// ---------------------------------------------------------------- kernel 4: q / alpha / yh
__global__ void k_qalpha(const float* __restrict__ xin,
                         const float* __restrict__ xp,
                         const float* __restrict__ theta,
                         const float* __restrict__ Atm,
                         const float* __restrict__ scale_p,
                         float* __restrict__ yhout) {
  __shared__ __align__(16) float XF[FUT * OPAD];
  __shared__ float VF[FUT * OPAD];
  __shared__ float TJ[FUT];
  __shared__ float YS[PAST];
  __shared__ float AL[FUT * PAST];
  const int b   = blockIdx.x;
  const int tid = threadIdx.x;   // blockDim = 64
  const float* xb = xin + (size_t)b * NPTS * OPAD;

  // async stage xf rows (cols 50/51 are never read)
  for (int e = tid * 4; e < FUT * OPAD; e += 64 * 4)
    async_g2l_b128(xb + PAST * OPAD + e, XF + e);
  YS[tid] = xb[tid * OPAD + 50];
  wait_async();
  __syncthreads();

  // VF[j][q] = sum_p XF[j][p] * theta[p][q]   (Theta symmetric)
  for (int e = tid; e < FUT * OPAD; e += 64) {
    int j = e / OPAD, q = e % OPAD;
    float s = 0.f;
    if (q < OFEAT)
      for (int p = 0; p < OFEAT; ++p) s += XF[j * OPAD + p] * theta[p * OPAD + q];
    VF[e] = s;
  }
  __syncthreads();

  if (tid < FUT) {
    float s = 0.f;
    for (int q = 0; q < OFEAT; ++q) s += VF[tid * OPAD + q] * XF[tid * OPAD + q];
    TJ[tid] = s;
  }
  __syncthreads();

  const float scale = scale_p[0];
  {
    const int i = tid;  // 0..63
    const float* xpi = xp + ((size_t)b * PAST + i) * OPAD;
    float xpr[OFEAT];
#pragma unroll
    for (int p = 0; p < OFEAT; ++p) xpr[p] = xpi[p];

    float si = 0.f;
    for (int q = 0; q < OFEAT; ++q) {
      float u = 0.f;
      for (int p = 0; p < OFEAT; ++p) u += xpr[p] * theta[p * OPAD + q];
      si += u * xpr[q];
    }

    float av[FUT], mx = NEGV;
#pragma unroll
    for (int j = 0; j < FUT; ++j) {
      float g = 0.f;
      for (int q = 0; q < OFEAT; ++q) g += VF[j * OPAD + q] * xpr[q];
      float qv = si + TJ[j] - 2.f * g;
      float a  = scale * qv;
      a = (Atm[j * PAST + i] > 0.5f) ? a : NEGV;
      av[j] = a;
      mx = fmaxf(mx, a);
    }
    float sum = 0.f;
#pragma unroll
    for (int j = 0; j < FUT; ++j) { av[j] = expf(av[j] - mx); sum += av[j]; }
    float inv = 1.f / sum;
#pragma unroll
    for (int j = 0; j < FUT; ++j) AL[j * PAST + i] = av[j] * inv;
  }
  __syncthreads();

  if (tid < FUT) {
    float s = 0.f;
    for (int p = 0; p < PAST; ++p) s += AL[tid * PAST + p] * YS[p];
    yhout[(size_t)b * FUT + tid] = s;
  }
}

// ---------------------------------------------------------------- kernel 5: GCNN #2 + out
__global__ void k_gcnn2(const float* __restrict__ xin,
                        const float* __restrict__ yh,
                        const float* __restrict__ An22,
                        const float* __restrict__ Wg2a,
                        const float* __restrict__ Wg2b,
                        float* __restrict__ outp) {
  __shared__ __align__(16) float X2[FUT * OPAD];
  __shared__ float T2[FUT * HID2];
  __shared__ float U2[FUT * HID2];
  __shared__ float t2s[FUT];
  const int b    = blockIdx.x;
  const int lane = threadIdx.x;  // blockDim = 32
  const float* xb = xin + (size_t)b * NPTS * OPAD;

  // async stage raw rows, then patch col 50 with yh (col 51 already 0)
  for (int e = lane * 4; e < FUT * OPAD; e += 32 * 4)
    async_g2l_b128(xb + PAST * OPAD + e, X2 + e);
  wait_async();
  __syncthreads();
  if (lane < FUT) X2[lane * OPAD + 50] = yh[(size_t)b * FUT + lane];
  __syncthreads();

  // T2 = X2 * Wg2a^T  (16x52)x(52x64)
  for (int nt = 0; nt < 4; ++nt) {
    int col = nt * 16 + (lane & 15);
    v8f c = splat8(0.f);
    for (int ks = 0; ks < 13; ++ks) {
      v2f a = afrag(X2, OPAD, 0, ks * 4);
      int k = ks * 4 + ((lane >> 4) << 1);
      v2f bb;
      bb[0] = (k + 0 < OUTP) ? Wg2a[col * OUTP + k + 0] : 0.f;
      bb[1] = (k + 1 < OUTP) ? Wg2a[col * OUTP + k + 1] : 0.f;
      c = wmma4(a, bb, c);
    }
#pragma unroll
    for (int r = 0; r < 8; ++r)
      T2[(r + ((lane >> 4) << 3)) * HID2 + col] = c[r];
  }
  __syncthreads();

  // U2 = relu(An22 * T2)  (16x16)x(16x64)
  for (int nt = 0; nt < 4; ++nt) {
    int col = nt * 16 + (lane & 15);
    v8f c = splat8(0.f);
    for (int ks = 0; ks < 4; ++ks) {
      v2f a  = afrag(An22, FUT, 0, ks * 4);
      v2f bb = bfrag(T2, HID2, ks * 4, nt * 16);
      c = wmma4(a, bb, c);
    }
#pragma unroll
    for (int r = 0; r < 8; ++r)
      U2[(r + ((lane >> 4) << 3)) * HID2 + col] = fmaxf(c[r], 0.f);
  }
  __syncthreads();

  // scalar tail: t2 = U2 @ Wg2b^T (16x1), out = An22 @ t2
  if (lane < FUT) {
    float s = 0.f;
    for (int k = 0; k < HID2; ++k) s += U2[lane * HID2 + k] * Wg2b[k];
    t2s[lane] = s;
  }
  __syncthreads();
  if (lane < FUT) {
    float o = 0.f;
    for (int p = 0; p < FUT; ++p) o += An22[lane * FUT + p] * t2s[p];
    outp[(size_t)b * FUT + lane] = o;
  }
}

// ---------------------------------------------------------------- kernel 6: pairwise distance
__global__ void k_dist(const float* __restrict__ xin,
                       const float* __restrict__ yh,
                       float* __restrict__ distout) {
  __shared__ __align__(16) float XC[NPTS * OPAD];
  __shared__ float SQ[NPTS];
  const int b    = blockIdx.x;
  const int tid  = threadIdx.x;  // blockDim = 64
  const int wave = tid >> 5;
  const int lane = tid & 31;
  const float* xb = xin + (size_t)b * NPTS * OPAD;

  // rows 0..63: raw async copy (col 51 zero in global x)
  for (int e = tid * 4; e < PAST * OPAD; e += 64 * 4)
    async_g2l_b128(xb + e, XC + e);
  // rows 64..79: merge yh into col 50
  for (int e = PAST * OPAD + tid; e < NPTS * OPAD; e += 64) {
    int n = e / OPAD, c = e % OPAD;
    float v;
    if (c >= OUTP)      v = 0.f;
    else if (c < OFEAT) v = xb[n * OPAD + c];
    else                v = yh[(size_t)b * FUT + (n - PAST)];  // c == 50
    XC[e] = v;
  }
  wait_async();
  __syncthreads();

  for (int n = tid; n < NPTS; n += 64) {
    float s = 0.f;
    for (int c = 0; c < OPAD; ++c) { float v = XC[n * OPAD + c]; s += v * v; }
    SQ[n] = s;
  }
  __syncthreads();

  for (int t = wave; t < 25; t += 2) {
    int mi = t / 5, ni = t % 5;
    int col = ni * 16 + (lane & 15);
    v8f c = splat8(0.f);
    for (int ks = 0; ks < 13; ++ks) {
      v2f a = afrag(XC, OPAD, mi * 16, ks * 4);
      int k = ks * 4 + ((lane >> 4) << 1);
      v2f bb;                               // B[k][n] = XC[n][k]
      bb[0] = XC[col * OPAD + k + 0];
      bb[1] = XC[col * OPAD + k + 1];
      c = wmma4(a, bb, c);
    }
#pragma unroll
    for (int r = 0; r < 8; ++r) {
      int m  = mi * 16 + r + ((lane >> 4) << 3);
      float d2 = SQ[m] + SQ[col] - 2.f * c[r];
      float v  = (d2 > 0.f) ? sqrtf(d2) : 0.f;
      distout[(size_t)b * NPTS * NPTS + m * NPTS + col] = v;
    }
  }
}

// ---------------------------------------------------------------- launch
extern "C" void kernel_launch(void* const* d_in, const int* in_sizes, int n_in,
                              void* d_out, int out_size, void* d_ws, size_t ws_size,
                              hipStream_t stream) {
  const float* data  = (const float*)d_in[0];
  const float* emb0  = (const float*)d_in[1];
  const float* emb1  = (const float*)d_in[2];
  const float* emb2  = (const float*)d_in[3];
  const float* w1    = (const float*)d_in[4];
  const float* b1    = (const float*)d_in[5];
  const float* w2    = (const float*)d_in[6];
  const float* b2    = (const float*)d_in[7];
  const float* w3    = (const float*)d_in[8];
  const float* b3    = (const float*)d_in[9];
  const float* Mg    = (const float*)d_in[10];
  const float* Wg1a  = (const float*)d_in[11];
  const float* Wg1b  = (const float*)d_in[12];
  const float* Wk    = (const float*)d_in[13];
  const float* smth  = (const float*)d_in[14];
  const float* Wg2a  = (const float*)d_in[15];
  const float* Wg2b  = (const float*)d_in[16];

  float* out  = (float*)d_out;                         // (B,16)
  float* dist = out + (size_t)BATCH * FUT;             // (B,80,80)
  float* Aout = dist + (size_t)BATCH * NPTS * NPTS;    // (80,80)

  float* ws = (float*)d_ws;
  size_t o = 0;
  float* x     = ws + o; o += (size_t)BATCH * NPTS * OPAD;  // (B,80,52)
  float* xp    = ws + o; o += (size_t)BATCH * PAST * OPAD;  // (B,64,52)
  float* yh    = ws + o; o += (size_t)BATCH * FUT;          // (B,16)
  float* An11  = ws + o; o += PAST * PAST;
  float* An22  = ws + o; o += FUT * FUT;
  float* Atm   = ws + o; o += FUT * PAST;
  float* theta = ws + o; o += OPAD * OPAD;
  float* scale = ws + o; o += 4;

  k_precompute<<<1, 128, 0, stream>>>(Mg, Wk, smth, Aout, An11, An22, Atm, theta, scale);
  k_mlp<<<(BATCH * NPTS) / 32, 64, 0, stream>>>(data, emb0, emb1, emb2,
                                                w1, b1, w2, b2, w3, b3, x);
  k_gcnn1<<<BATCH, 64, 0, stream>>>(x, An11, Wg1a, Wg1b, xp);
  k_qalpha<<<BATCH, 64, 0, stream>>>(x, xp, theta, Atm, scale, yh);
  k_gcnn2<<<BATCH, 32, 0, stream>>>(x, yh, An22, Wg2a, Wg2b, out);
  k_dist<<<BATCH, 64, 0, stream>>>(x, yh, dist);
}